// ReconStep_7842610283461
// MI455X (gfx1250) — compile-verified
//
#include <hip/hip_runtime.h>
#include <math.h>

#define S_SAMPLES 96
#define EPS_F 1e-8f
// KW = sqrt(3*3/pi) = 3/sqrt(pi)
#define KW_F 1.6925687506432687f

typedef __attribute__((ext_vector_type(2))) float v2f;
typedef __attribute__((ext_vector_type(4))) float v4f;
typedef __attribute__((ext_vector_type(8))) float v8f;

// Full 32-lane sum, broadcast to all lanes, via the CDNA5 matrix path.
// A (16x4 f32, documented layout): lane L<16 holds A[L,0] in VGPR0; lane
// L>=16 holds A[L-16,2]. Partials go in the VGPR0 slot, zeros in VGPR1;
// B = all ones (layout-independent). Then D[m,n] = p_m + p_{m+16} =: q_m for
// every n, and D VGPR r carries q_r (lanes 0-15) / q_{r+8} (lanes 16-31).
// Summing the 8 D registers gives each half-wave its half-sum; one
// ds_swizzle SWAPX16 (offset 0x401F) completes the reduction.
__device__ __forceinline__ float wave32_sum(float v) {
  v2f a; a.x = v;    a.y = 0.0f;
  v2f b; b.x = 1.0f; b.y = 1.0f;
  v8f c = {};
  c = __builtin_amdgcn_wmma_f32_16x16x4_f32(false, a, false, b, (short)0, c,
                                            false, false);
  float h = ((c[0] + c[1]) + (c[2] + c[3])) + ((c[4] + c[5]) + (c[6] + c[7]));
  float o = __int_as_float(
      __builtin_amdgcn_ds_swizzle(__float_as_int(h), 0x401f));
  return h + o;
}

// Fused project + backproject for one LOR handled by one wave32
// (96 samples = 3 per lane). Strides (s0,s1,s2) fold the reference's
// image/backprojection transposes into permuted indexing of the canonical
// z-layout volume; gather and scatter share them. Entirely branch-free:
// OOB corners use clamped addresses with zeroed separable weights, exactly
// matching the reference's clip + where(w,0).
__device__ __forceinline__ void process_lor(
    const float* __restrict__ img, float* __restrict__ acc,
    const float* __restrict__ Lp, int lane, int D,
    int s0, int s1, int s2,
    float ox, float oy, float oz, float ivx, float ivy, float ivz)
{
  const float p1x = Lp[0], p1y = Lp[1], p1z = Lp[2];
  const float dx = Lp[3] - p1x, dy = Lp[4] - p1y, dz = Lp[5] - p1z;
  const float seg = sqrtf(dx*dx + dy*dy + dz*dz) * (1.0f / S_SAMPLES);

  int   ibx[3], iby[3], ibz[3];
  float gfx[3], gfy[3], gfz[3];
  float partial = 0.0f;

  #pragma unroll
  for (int j = 0; j < 3; ++j) {
    const float t  = ((float)(lane + 32*j) + 0.5f) * (1.0f / S_SAMPLES);
    const float vx = (fmaf(t, dx, p1x) + ox) * ivx - 0.5f;
    const float vy = (fmaf(t, dy, p1y) + oy) * ivy - 0.5f;
    const float vz = (fmaf(t, dz, p1z) + oz) * ivz - 0.5f;
    const float fx = floorf(vx), fy = floorf(vy), fz = floorf(vz);
    const int   ix = (int)fx,    iy = (int)fy,    iz = (int)fz;
    const float gx = vx - fx,    gy = vy - fy,    gz = vz - fz;
    ibx[j] = ix; iby[j] = iy; ibz[j] = iz;
    gfx[j] = gx; gfy[j] = gy; gfz[j] = gz;

    // clamped per-axis indices + OOB-masked separable weights
    const int x0 = min(max(ix, 0), D-1),   x1 = min(max(ix+1, 0), D-1);
    const int y0 = min(max(iy, 0), D-1),   y1 = min(max(iy+1, 0), D-1);
    const int z0 = min(max(iz, 0), D-1),   z1 = min(max(iz+1, 0), D-1);
    const float wx0 = ((unsigned)ix     < (unsigned)D) ? (1.0f-gx) : 0.0f;
    const float wx1 = ((unsigned)(ix+1) < (unsigned)D) ? gx        : 0.0f;
    const float wy0 = ((unsigned)iy     < (unsigned)D) ? (1.0f-gy) : 0.0f;
    const float wy1 = ((unsigned)(iy+1) < (unsigned)D) ? gy        : 0.0f;
    const float wz0 = ((unsigned)iz     < (unsigned)D) ? (1.0f-gz) : 0.0f;
    const float wz1 = ((unsigned)(iz+1) < (unsigned)D) ? gz        : 0.0f;

    const int xs0 = x0*s0, xs1 = x1*s0;
    const int ys0 = y0*s1, ys1 = y1*s1;
    const int zs0 = z0*s2, zs1 = z1*s2;
    const int a00 = xs0+ys0, a01 = xs0+ys1, a10 = xs1+ys0, a11 = xs1+ys1;

    // 8 unconditional gathers (valid clamped addresses) -> deep MLP
    const float v000 = img[a00+zs0], v001 = img[a00+zs1];
    const float v010 = img[a01+zs0], v011 = img[a01+zs1];
    const float v100 = img[a10+zs0], v101 = img[a10+zs1];
    const float v110 = img[a11+zs0], v111 = img[a11+zs1];

    const float w00 = wx0*wy0, w01 = wx0*wy1, w10 = wx1*wy0, w11 = wx1*wy1;
    partial += w00*(wz0*v000 + wz1*v001) + w01*(wz0*v010 + wz1*v011)
             + w10*(wz0*v100 + wz1*v101) + w11*(wz0*v110 + wz1*v111);
  }

  // line integral = total * seg * KW; scatter value = w * LI * seg * KW
  const float total = wave32_sum(partial);   // EXEC all-ones here
  const float sk    = seg * KW_F;
  const float cval  = total * sk * sk;

  #pragma unroll
  for (int j = 0; j < 3; ++j) {
    const int   ix = ibx[j], iy = iby[j], iz = ibz[j];
    const float gx = gfx[j], gy = gfy[j], gz = gfz[j];
    const int x0 = min(max(ix, 0), D-1),   x1 = min(max(ix+1, 0), D-1);
    const int y0 = min(max(iy, 0), D-1),   y1 = min(max(iy+1, 0), D-1);
    const int z0 = min(max(iz, 0), D-1),   z1 = min(max(iz+1, 0), D-1);
    const float wx0 = ((unsigned)ix     < (unsigned)D) ? (1.0f-gx) : 0.0f;
    const float wx1 = ((unsigned)(ix+1) < (unsigned)D) ? gx        : 0.0f;
    const float wy0 = ((unsigned)iy     < (unsigned)D) ? (1.0f-gy) : 0.0f;
    const float wy1 = ((unsigned)(iy+1) < (unsigned)D) ? gy        : 0.0f;
    const float wz0 = ((unsigned)iz     < (unsigned)D) ? (1.0f-gz) : 0.0f;
    const float wz1 = ((unsigned)(iz+1) < (unsigned)D) ? gz        : 0.0f;

    const int xs0 = x0*s0, xs1 = x1*s0;
    const int ys0 = y0*s1, ys1 = y1*s1;
    const int zs0 = z0*s2, zs1 = z1*s2;
    const int a00 = xs0+ys0, a01 = xs0+ys1, a10 = xs1+ys0, a11 = xs1+ys1;

    const float w00 = wx0*wy0*cval, w01 = wx0*wy1*cval;
    const float w10 = wx1*wy0*cval, w11 = wx1*wy1*cval;
    atomicAdd(&acc[a00+zs0], w00*wz0);
    atomicAdd(&acc[a00+zs1], w00*wz1);
    atomicAdd(&acc[a01+zs0], w01*wz0);
    atomicAdd(&acc[a01+zs1], w01*wz1);
    atomicAdd(&acc[a10+zs0], w10*wz0);
    atomicAdd(&acc[a10+zs1], w10*wz1);
    atomicAdd(&acc[a11+zs0], w11*wz0);
    atomicAdd(&acc[a11+zs1], w11*wz1);
  }
}

// Specialized kernel: compile-time D and strides (one instantiation per
// LOR set) -> index math folds to shifts/immediates, no set branches.
template<int D, int S0, int S1, int S2>
__global__ void __launch_bounds__(128)
pet_fused_tmpl(const float* __restrict__ img, const float* __restrict__ lors,
               float* __restrict__ acc, const float* __restrict__ center3,
               const float* __restrict__ size3, int N)
{
  const int lane = threadIdx.x & 31;
  const int lor  = blockIdx.x * (blockDim.x >> 5) + (threadIdx.x >> 5);
  if (lor >= N) return;                      // wave-uniform exit
  const float sx = size3[0], sy = size3[1], sz = size3[2];
  const float ox = 0.5f*sx - center3[0];
  const float oy = 0.5f*sy - center3[1];
  const float oz = 0.5f*sz - center3[2];
  const float* Lp = lors + (size_t)lor * 6;
  __builtin_prefetch(Lp, 0, 1);
  process_lor(img, acc, Lp, lane, D, S0, S1, S2, ox, oy, oz,
              (float)D/sx, (float)D/sy, (float)D/sz);
}

// Generic fallback for arbitrary D (runtime strides, set from blockIdx.y).
__global__ void __launch_bounds__(128)
pet_fused_gen(const float* __restrict__ img,
              const float* __restrict__ zl, const float* __restrict__ xl,
              const float* __restrict__ yl, float* __restrict__ acc,
              const float* __restrict__ center3,
              const float* __restrict__ size3, int N, int D)
{
  const int lane = threadIdx.x & 31;
  const int lor  = blockIdx.x * (blockDim.x >> 5) + (threadIdx.x >> 5);
  if (lor >= N) return;
  const int set = blockIdx.y;
  const float* lors = (set == 0) ? zl : (set == 1) ? xl : yl;
  const int DD = D * D;
  int s0, s1, s2;
  if (set == 0)      { s0 = DD; s1 = D;  s2 = 1; } // z: (ix,iy,iz)
  else if (set == 1) { s0 = 1;  s1 = DD; s2 = D; } // x: (iy,iz,ix)
  else               { s0 = D;  s1 = DD; s2 = 1; } // y: (iy,ix,iz)
  const float sx = size3[0], sy = size3[1], sz = size3[2];
  const float ox = 0.5f*sx - center3[0];
  const float oy = 0.5f*sy - center3[1];
  const float oz = 0.5f*sz - center3[2];
  const float* Lp = lors + (size_t)lor * 6;
  __builtin_prefetch(Lp, 0, 1);
  process_lor(img, acc, Lp, lane, D, s0, s1, s2, ox, oy, oz,
              (float)D/sx, (float)D/sy, (float)D/sz);
}

// out = image / (eff + EPS) * accumulated_backprojection, 4-wide streaming.
__global__ void __launch_bounds__(256)
pet_finalize_kernel(const float* __restrict__ img,
                    const float* __restrict__ eff,
                    float* __restrict__ out, int n4)
{
  const int i = blockIdx.x * blockDim.x + threadIdx.x;
  if (i >= n4) return;
  v4f a = ((const v4f*)img)[i];
  v4f e = __builtin_nontemporal_load((const v4f*)eff + i);
  v4f o = ((const v4f*)out)[i];
  v4f r;
  r.x = a.x / (e.x + EPS_F) * o.x;
  r.y = a.y / (e.y + EPS_F) * o.y;
  r.z = a.z / (e.z + EPS_F) * o.z;
  r.w = a.w / (e.w + EPS_F) * o.w;
  ((v4f*)out)[i] = r;
}

extern "C" void kernel_launch(void* const* d_in, const int* in_sizes, int n_in,
                              void* d_out, int out_size, void* d_ws,
                              size_t ws_size, hipStream_t stream)
{
  const float* img    = (const float*)d_in[0];
  const float* eff    = (const float*)d_in[1];
  // d_in[2] = grid (int32[3]) — dims recovered from in_sizes instead
  const float* center = (const float*)d_in[3];
  const float* fov    = (const float*)d_in[4];
  const float* xl     = (const float*)d_in[5];
  const float* yl     = (const float*)d_in[6];
  const float* zl     = (const float*)d_in[7];
  float* out = (float*)d_out;

  const int N = in_sizes[5] / 6;
  int D = 1;
  while ((long long)D * D * D < (long long)in_sizes[0]) ++D;  // 192

  // Zero accumulator every call (deterministic; capture-safe memset node).
  hipMemsetAsync(out, 0, (size_t)out_size * sizeof(float), stream);

  const int wavesPerBlock = 4;                  // 128 threads = 4 wave32
  const int nblk = (N + wavesPerBlock - 1) / wavesPerBlock;

  if (D == 192) {
    constexpr int DC = 192, DD = 192 * 192;
    pet_fused_tmpl<DC, DD, DC, 1><<<nblk, 128, 0, stream>>>(
        img, zl, out, center, fov, N);            // z-set
    pet_fused_tmpl<DC, 1, DD, DC><<<nblk, 128, 0, stream>>>(
        img, xl, out, center, fov, N);            // x-set
    pet_fused_tmpl<DC, DC, DD, 1><<<nblk, 128, 0, stream>>>(
        img, yl, out, center, fov, N);            // y-set
  } else {
    dim3 grid(nblk, 3, 1);
    pet_fused_gen<<<grid, 128, 0, stream>>>(img, zl, xl, yl, out, center,
                                            fov, N, D);
  }

  const int n4 = out_size / 4;                    // 192^3 divisible by 4
  pet_finalize_kernel<<<(n4 + 255) / 256, 256, 0, stream>>>(img, eff, out, n4);
}